// Gate_8469675508071
// MI455X (gfx1250) — compile-verified
//
#include <hip/hip_runtime.h>
#include <hip/hip_bf16.h>
#include <stdint.h>

typedef __attribute__((ext_vector_type(2))) float v2f;
typedef __attribute__((ext_vector_type(8))) float v8f;
typedef int v4i_vs __attribute__((vector_size(16)));   // matches builtin param type

#define HID    2048
#define NEXP   256
#define BM     128
#define BN     128
#define BK     32
#define BKP    36   // padded LDS stride: 144 B (16B-aligned float4 stores, conflict-free frag reads)

#if __has_builtin(__builtin_amdgcn_global_load_async_to_lds_b128)
#define HAS_ASYNC_LDS 1
#else
#define HAS_ASYNC_LDS 0
#endif

// stage one float4 from global to LDS (async-to-LDS when available)
__device__ __forceinline__ void stage16(const float* __restrict__ g, float* l) {
#if HAS_ASYNC_LDS
    __builtin_amdgcn_global_load_async_to_lds_b128(
        (v4i_vs*)(g), (v4i_vs*)(l), /*offset=*/0, /*cpol=*/0);
#else
    *reinterpret_cast<float4*>(l) = *reinterpret_cast<const float4*>(g);
#endif
}

__device__ __forceinline__ void wait_async_le8() {
#if HAS_ASYNC_LDS
#if __has_builtin(__builtin_amdgcn_s_wait_asynccnt)
    __builtin_amdgcn_s_wait_asynccnt(8);
#else
    asm volatile("s_wait_asynccnt 8" ::: "memory");
#endif
#endif
}

// ---------------------------------------------------------------------------
// Kernel 1: scores[t][e] = sigmoid( x[t,:] . W[e,:] + bias[e] )
// Block = 256 threads = 8 waves laid out 4(M) x 2(N); block tile 128M x 128N.
// Each wave owns a 32(M) x 64(N) strip => eight 16x16 f32 accumulators,
// fed by V_WMMA_F32_16X16X4_F32 over K=2048. A frags reused 4x, amortized
// LDS traffic ~6B/lane/WMMA. Double-buffered async-to-LDS staging.
// ---------------------------------------------------------------------------
__global__ __launch_bounds__(256, 2)
void score_gemm_kernel(const float* __restrict__ x,
                       const float* __restrict__ w,
                       const float* __restrict__ bias,
                       float* __restrict__ scores)
{
    __shared__ float As[2][BM][BKP];   // x tile    (128 x 32 used)  2 x 18 KB
    __shared__ float Bs[2][BN][BKP];   // W tile    (128 x 32 used)  2 x 18 KB

    const int tid  = threadIdx.x;
    const int lane = tid & 31;
    const int wave = tid >> 5;
    const int wm   = wave >> 1;        // 0..3  (M sub-tile, 32 rows each)
    const int wn   = wave & 1;         // 0..1  (N sub-tile, 64 cols each)

    const int m0 = blockIdx.x * BM;    // token base
    const int n0 = blockIdx.y * BN;    // expert base

    const int khi = (lane >> 4) << 1;  // lanes 0-15 -> K {0,1}, lanes 16-31 -> K {2,3}
    const int l16 = lane & 15;

    v8f acc[2][4];
    #pragma unroll
    for (int mi = 0; mi < 2; ++mi)
        #pragma unroll
        for (int t = 0; t < 4; ++t)
            acc[mi][t] = (v8f){0.f,0.f,0.f,0.f,0.f,0.f,0.f,0.f};

    // cooperative staging map: 128 rows x 32 cols = 4096 floats per tile,
    // 16 floats (4x float4) per thread; thread -> (row = tid/2, cols = (tid&1)*16 ..+15)
    const int srow = tid >> 1;          // 0..127
    const int scol = (tid & 1) * 16;    // 0 or 16

    auto issue_tiles = [&](int buf, int kb) {
        const float* ga = x + (size_t)(m0 + srow) * HID + kb + scol;
        const float* gb = w + (size_t)(n0 + srow) * HID + kb + scol;
        #pragma unroll
        for (int i = 0; i < 4; ++i) {
            stage16(ga + i * 4, &As[buf][srow][scol + i * 4]);
            stage16(gb + i * 4, &Bs[buf][srow][scol + i * 4]);
        }
    };

    constexpr int NITER = HID / BK;    // 64
    issue_tiles(0, 0);

    for (int it = 0; it < NITER; ++it) {
        const int buf = it & 1;
        // prefetch next K-block (clamped repeat on the last iteration keeps
        // control flow uniform and addresses in-bounds)
        const int kbn = (it + 1 < NITER) ? (it + 1) * BK : it * BK;
        issue_tiles(buf ^ 1, kbn);

        wait_async_le8();              // our 8 oldest (current tile) have landed
        __syncthreads();               // everyone's portion visible

        #pragma unroll
        for (int kk = 0; kk < BK; kk += 4) {
            v2f a[2];
            #pragma unroll
            for (int mi = 0; mi < 2; ++mi) {
                a[mi].x = As[buf][wm * 32 + mi * 16 + l16][kk + khi];
                a[mi].y = As[buf][wm * 32 + mi * 16 + l16][kk + khi + 1];
            }
            #pragma unroll
            for (int t = 0; t < 4; ++t) {
                const int n = wn * 64 + t * 16 + l16;
                v2f b;
                b.x = Bs[buf][n][kk + khi];
                b.y = Bs[buf][n][kk + khi + 1];
                #pragma unroll
                for (int mi = 0; mi < 2; ++mi) {
                    acc[mi][t] = __builtin_amdgcn_wmma_f32_16x16x4_f32(
                        /*neg_a=*/false, a[mi], /*neg_b=*/false, b,
                        /*c_mod=*/(short)0, acc[mi][t],
                        /*reuse_a=*/false, /*reuse_b=*/false);
                }
            }
        }
        __syncthreads();               // done reading buf before it is refilled
    }

    // epilogue: + bias, sigmoid, store. C layout: VGPR r -> M = r + 8*(lane/16); N = lane%16.
    #pragma unroll
    for (int mi = 0; mi < 2; ++mi) {
        #pragma unroll
        for (int t = 0; t < 4; ++t) {
            const int e  = n0 + wn * 64 + t * 16 + l16;
            const float be = bias[e];
            #pragma unroll
            for (int r = 0; r < 8; ++r) {
                const int tok = m0 + wm * 32 + mi * 16 + ((lane >> 4) << 3) + r;
                const float v = acc[mi][t][r] + be;
                scores[(size_t)tok * NEXP + e] = 1.0f / (1.0f + __expf(-v));
            }
        }
    }
}

// ---------------------------------------------------------------------------
// Kernel 2: grouped top-k routing. One wave32 per token; lane l owns expert
// g*32+l of each of the 8 groups. 8 tokens per 256-thread block.
// ---------------------------------------------------------------------------
#define NEG_INF (-3.402823466e38f)

__global__ __launch_bounds__(256)
void route_kernel(const float* __restrict__ scores,
                  const float* __restrict__ bias,
                  float* __restrict__ w_out,
                  int* __restrict__ idx_out,
                  int n_tokens)
{
    const int lane  = threadIdx.x & 31;
    const int wave  = threadIdx.x >> 5;
    const int token = blockIdx.x * 8 + wave;
    if (token >= n_tokens) return;

    const float* srow = scores + (size_t)token * NEXP;

    float sorig[8], sb[8];
    #pragma unroll
    for (int g = 0; g < 8; ++g) {
        const int e = g * 32 + lane;
        sorig[g] = srow[e];
        sb[g]    = sorig[g] + bias[e];
    }

    // group score = sum of top-2 of each group's 32 lane-values
    float gs[8];
    #pragma unroll
    for (int g = 0; g < 8; ++g) {
        float hi = sb[g], lo = NEG_INF;
        #pragma unroll
        for (int off = 16; off >= 1; off >>= 1) {
            const float ohi = __shfl_xor(hi, off, 32);
            const float olo = __shfl_xor(lo, off, 32);
            const float nh = fmaxf(hi, ohi);
            const float nl = fmaxf(fminf(hi, ohi), fmaxf(lo, olo));
            hi = nh; lo = nl;
        }
        gs[g] = hi + lo;   // uniform across the wave
    }

    // top-4 groups, stable tie-break (lower group index wins, like jax top_k)
    float ms[8];
    #pragma unroll
    for (int g = 0; g < 8; ++g) {
        int rank = 0;
        #pragma unroll
        for (int h = 0; h < 8; ++h)
            rank += (gs[h] > gs[g]) || (gs[h] == gs[g] && h < g);
        ms[g] = (rank < 4) ? sb[g] : NEG_INF;
    }

    // iteratively extract top-8 experts over the masked biased scores
    float picked_w[8];
    int   picked_e[8];
    #pragma unroll
    for (int i = 0; i < 8; ++i) {
        float bv = NEG_INF, bo = 0.0f;
        int   be = 0x7FFFFFFF;
        #pragma unroll
        for (int g = 0; g < 8; ++g) {
            const int e = g * 32 + lane;
            if (ms[g] > bv || (ms[g] == bv && e < be)) {
                bv = ms[g]; be = e; bo = sorig[g];
            }
        }
        #pragma unroll
        for (int off = 16; off >= 1; off >>= 1) {
            const float ov = __shfl_xor(bv, off, 32);
            const int   oe = __shfl_xor(be, off, 32);
            const float oo = __shfl_xor(bo, off, 32);
            if (ov > bv || (ov == bv && oe < be)) { bv = ov; be = oe; bo = oo; }
        }
        picked_w[i] = bo;
        picked_e[i] = be;
        if ((be & 31) == lane) ms[be >> 5] = NEG_INF;  // owning lane retires it
    }

    float sum = 0.0f;
    #pragma unroll
    for (int i = 0; i < 8; ++i) sum += picked_w[i];
    const float scale = 2.5f / sum;

    if (lane == 0) {
        #pragma unroll
        for (int i = 0; i < 8; ++i) {
            w_out[(size_t)token * 8 + i]   = picked_w[i] * scale;
            idx_out[(size_t)token * 8 + i] = picked_e[i];
        }
    }
}

// ---------------------------------------------------------------------------
extern "C" void kernel_launch(void* const* d_in, const int* in_sizes, int n_in,
                              void* d_out, int out_size, void* d_ws, size_t ws_size,
                              hipStream_t stream)
{
    const float* x    = (const float*)d_in[0];   // [N_TOKENS, 2048]
    const float* w    = (const float*)d_in[1];   // [256, 2048]
    const float* bias = (const float*)d_in[2];   // [256]

    const int n_tokens = in_sizes[0] / HID;      // 16384

    float* scores = (float*)d_ws;                // [n_tokens, 256] fp32 (16 MB)

    float* w_out  = (float*)d_out;               // weights [n_tokens, 8] fp32
    int*   i_out  = (int*)d_out + (size_t)n_tokens * 8;  // indices [n_tokens, 8] int32

    dim3 gemm_grid(n_tokens / BM, NEXP / BN);
    score_gemm_kernel<<<gemm_grid, 256, 0, stream>>>(x, w, bias, scores);

    dim3 route_grid((n_tokens + 7) / 8);
    route_kernel<<<route_grid, 256, 0, stream>>>(scores, bias, w_out, i_out, n_tokens);
}